// Model_74010876445396
// MI455X (gfx1250) — compile-verified
//
#include <hip/hip_runtime.h>

// out[b] = dot(x[b,:], colsum(W)) * COMBINED_SCALE, COMBINED_SCALE = 2.0/2.0 = 1.0
#define COMBINED_SCALE 1.0f
#define IN_DIM  4096   // i (columns of W, columns of x)
#define HID_DIM 4096   // h (rows of W)
#define BATCH   32768

typedef __attribute__((ext_vector_type(2))) float v2f;
typedef __attribute__((ext_vector_type(8))) float v8f;

// ---------------------------------------------------------------------------
// Stage 1: partial column-sums of W via V_WMMA_F32_16X16X4_F32.
//   C(16x16) += A(16x4 of ones) x B(4x16 tile of W)
// Each wave owns a 16-column strip; gridDim.y splits the 4096 rows into 8
// chunks of 512. Block = 256 threads = 8 waves -> 128 columns per block.
// grid = (4096/128, 8) = (32, 8).
// partial layout: partial[chunk * 4096 + col]
// ---------------------------------------------------------------------------
__global__ void __launch_bounds__(256)
colsum_wmma_kernel(const float* __restrict__ W, float* __restrict__ partial)
{
    const int lane = threadIdx.x & 31;
    const int wave = threadIdx.x >> 5;
    const int col  = blockIdx.x * 128 + wave * 16 + (lane & 15);
    const int row0 = blockIdx.y * 512;
    // Lanes 0-15 carry K rows {r, r+1}, lanes 16-31 carry {r+2, r+3}.
    // (A is all-ones, so the K<->lane-half assignment cannot change the sum.)
    const int k_lo = (lane < 16) ? 0 : 2;

    v2f a; a.x = 1.0f; a.y = 1.0f;     // A = ones (layout-invariant)
    v8f c = {};                        // C accumulator = 0

    const float* base = W + (size_t)(row0 + k_lo) * IN_DIM + col;
    #pragma unroll 4
    for (int r = 0; r < 512; r += 4) {
        v2f b;
        b.x = base[(size_t)(r    ) * IN_DIM];
        b.y = base[(size_t)(r + 1) * IN_DIM];
        // 8 args: (neg_a, A, neg_b, B, c_mod, C, reuse_a, reuse_b)
        c = __builtin_amdgcn_wmma_f32_16x16x4_f32(
                false, a, false, b, (short)0, c, false, false);
    }

    // C/D 16x16 f32 layout: VGPR0, lanes 0-15 -> M=0, N=lane.
    if (lane < 16) {
        partial[(size_t)blockIdx.y * IN_DIM + col] = c[0];
    }
}

// ---------------------------------------------------------------------------
// Stage 2: deterministic sequential reduction of the 8 partial colsum chunks.
// grid = (16), block = 256 -> 4096 threads, one per column.
// ---------------------------------------------------------------------------
__global__ void __launch_bounds__(256)
reduce_partials_kernel(const float* __restrict__ partial, float* __restrict__ colsum)
{
    const int i = blockIdx.x * 256 + threadIdx.x;
    float s = 0.0f;
    #pragma unroll
    for (int j = 0; j < 8; ++j) s += partial[(size_t)j * IN_DIM + i];
    colsum[i] = s;
}

// ---------------------------------------------------------------------------
// Stage 3: out[b] = dot(x[b,:], colsum) * scale.
// One wave32 per row; block = 256 threads = 8 rows; grid = 32768/8 = 4096.
// colsum staged in LDS (16 KB) once per block; x streamed with b128 loads.
// ---------------------------------------------------------------------------
__global__ void __launch_bounds__(256)
matvec_kernel(const float* __restrict__ x, const float* __restrict__ colsum,
              float* __restrict__ out)
{
    __shared__ float s_sum[IN_DIM];
    for (int i = threadIdx.x; i < IN_DIM; i += 256) s_sum[i] = colsum[i];
    __syncthreads();

    const int lane = threadIdx.x & 31;
    const int wave = threadIdx.x >> 5;
    const int row  = blockIdx.x * 8 + wave;

    const float4* __restrict__ xr = (const float4*)(x + (size_t)row * IN_DIM);
    const float4* __restrict__ sr = (const float4*)s_sum;

    float acc = 0.0f;
    #pragma unroll 4
    for (int it = 0; it < IN_DIM / (32 * 4); ++it) {   // 32 iters
        const int idx = it * 32 + lane;
        float4 xv = xr[idx];
        float4 sv = sr[idx];
        acc = fmaf(xv.x, sv.x, acc);
        acc = fmaf(xv.y, sv.y, acc);
        acc = fmaf(xv.z, sv.z, acc);
        acc = fmaf(xv.w, sv.w, acc);
    }

    // wave32 reduction
    #pragma unroll
    for (int off = 16; off > 0; off >>= 1)
        acc += __shfl_down(acc, off, 32);

    if (lane == 0) out[row] = acc * COMBINED_SCALE;
}

// ---------------------------------------------------------------------------
extern "C" void kernel_launch(void* const* d_in, const int* in_sizes, int n_in,
                              void* d_out, int out_size, void* d_ws, size_t ws_size,
                              hipStream_t stream)
{
    (void)in_sizes; (void)n_in; (void)out_size; (void)ws_size;

    const float* x = (const float*)d_in[0];   // (32768, 4096)
    const float* W = (const float*)d_in[1];   // (4096, 4096)
    float* out     = (float*)d_out;           // (32768, 1)

    float* colsum  = (float*)d_ws;            // 4096 floats
    float* partial = colsum + IN_DIM;         // 8 * 4096 floats

    colsum_wmma_kernel     <<<dim3(32, 8), 256, 0, stream>>>(W, partial);
    reduce_partials_kernel <<<16,          256, 0, stream>>>(partial, colsum);
    matvec_kernel          <<<BATCH / 8,   256, 0, stream>>>(x, colsum, out);
}